// Pooling3D_53790170415478
// MI455X (gfx1250) — compile-verified
//
#include <hip/hip_runtime.h>
#include <hip/hip_bf16.h>

// ---- problem constants (match reference setup_inputs) ----
#define N_NODES 16384
#define N_EDGES 524288
#define D_FEAT  128
#define BATCH   4
#define NPTS    (N_NODES / BATCH)   // 4096 points per cloud
#define KSEL    (NPTS / 2)          // round(4096 * (1-0.5)) = 2048 sampled per cloud
#define M_SUB   (BATCH * KSEL)      // 8192 sampled nodes total

// ---- CDNA5 async global->LDS path (builtins confirmed present on this
//      toolchain in round 1: the guarded call itself produced the round-1
//      type diagnostic, so the __has_builtin guard passes) ----
#if defined(__has_builtin)
#if __has_builtin(__builtin_amdgcn_global_load_async_to_lds_b128) && \
    __has_builtin(__builtin_amdgcn_s_wait_asynccnt)
#define HAVE_ASYNC_LDS 1
#endif
#endif

#define AS1 __attribute__((address_space(1)))
#define AS3 __attribute__((address_space(3)))
typedef int v4i __attribute__((__vector_size__(4 * sizeof(int))));

// =====================================================================
// Kernel 1: init workspace + zero the sub_feat accumulator region
// =====================================================================
__global__ void pool_init_kernel(int* __restrict__ inv, float* __restrict__ deg,
                                 float* __restrict__ subfeat) {
  int i = blockIdx.x * blockDim.x + threadIdx.x;
  if (i < N_NODES) { inv[i] = -1; deg[i] = 0.0f; }
  if (i < M_SUB * D_FEAT) subfeat[i] = 0.0f;
}

// =====================================================================
// Kernel 2: farthest point sampling, one workgroup per cloud.
// Cloud xyz staged into LDS via async-to-LDS b128; each thread owns
// 4 points whose coords + running min-dist live in registers.
// Argmax per step: intra-wave shfl reduce, then one ds_max_u64 per wave
// leader into a rotating LDS slot -> ONE barrier per serial step.
// =====================================================================
__global__ __launch_bounds__(1024) void fps_kernel(
    const float* __restrict__ pos, int* __restrict__ inv,
    int* __restrict__ fpi, float* __restrict__ out_subx,
    float* __restrict__ out_fp) {
  __shared__ float p[NPTS * 3];                   // 48 KB: interleaved x,y,z
  __shared__ unsigned long long amax[3];          // rotating argmax slots

  const int b   = blockIdx.x;
  const int tid = threadIdx.x;
  const float* gpos = pos + (size_t)b * NPTS * 3;

  if (tid < 3) amax[tid] = 0ull;

#ifdef HAVE_ASYNC_LDS
  // 12288 floats = 3072 x b128 transfers; per-lane copy straight to LDS.
  for (int t = tid; t < (NPTS * 3) / 4; t += 1024) {
    __builtin_amdgcn_global_load_async_to_lds_b128(
        (AS1 v4i*)(gpos + t * 4), (AS3 v4i*)(p + t * 4), 0, 0);
  }
  __builtin_amdgcn_s_wait_asynccnt(0);
#else
  for (int t = tid; t < NPTS * 3; t += 1024) p[t] = gpos[t];
#endif
  __syncthreads();

  // own 4 points in registers (i = tid + j*1024)
  float px[4], py[4], pz[4], dmin[4];
#pragma unroll
  for (int j = 0; j < 4; ++j) {
    int i = tid + j * 1024;
    px[j] = p[i * 3 + 0];
    py[j] = p[i * 3 + 1];
    pz[j] = p[i * 3 + 2];
    dmin[j] = 1e10f;
  }

  int last = 0;
  if (tid == 0) {  // emit t=0: deterministic start at local index 0
    int m = b * KSEL, g = b * NPTS;
    inv[g] = m;
    fpi[m] = g;
    out_fp[m] = (float)g;
    out_subx[m * 3 + 0] = p[0];
    out_subx[m * 3 + 1] = p[1];
    out_subx[m * 3 + 2] = p[2];
  }

  int slot = 1;  // slot(t) = t % 3, starting at t = 1
  for (int t = 1; t < KSEL; ++t) {
    float qx = p[last * 3 + 0];   // LDS broadcast reads
    float qy = p[last * 3 + 1];
    float qz = p[last * 3 + 2];

    float bv = -1.0f;
    int   bi = tid;
#pragma unroll
    for (int j = 0; j < 4; ++j) {
      float dx = px[j] - qx, dy = py[j] - qy, dz = pz[j] - qz;
      float ds = dx * dx + dy * dy + dz * dz;
      dmin[j] = fminf(dmin[j], ds);
      if (dmin[j] > bv) { bv = dmin[j]; bi = tid + j * 1024; }
    }
    // wave32 argmax (tie-break: lowest index, matching jnp.argmax)
#pragma unroll
    for (int off = 16; off > 0; off >>= 1) {
      float ov = __shfl_xor(bv, off);
      int   oi = __shfl_xor(bi, off);
      if (ov > bv || (ov == bv && oi < bi)) { bv = ov; bi = oi; }
    }
    // packed key: dist bits (non-negative -> monotonic) in high word,
    // inverted index in low word so ties pick the lowest index.
    int nslot = slot + 1; if (nslot == 3) nslot = 0;
    if ((tid & 31) == 0) {
      unsigned long long key =
          ((unsigned long long)__float_as_uint(bv) << 32) |
          (unsigned int)(NPTS - 1 - bi);
      atomicMax(&amax[slot], key);
    }
    if (tid == 0) amax[nslot] = 0ull;  // reset slot(t+1); safe per 3-slot rotation
    __syncthreads();

    unsigned long long kk = amax[slot];
    int i2 = (NPTS - 1) - (int)(unsigned int)(kk & 0xFFFFFFFFull);
    if (tid == 0) {
      int m = b * KSEL + t, g = b * NPTS + i2;
      inv[g] = m;
      fpi[m] = g;
      out_fp[m] = (float)g;
      out_subx[m * 3 + 0] = p[i2 * 3 + 0];
      out_subx[m * 3 + 1] = p[i2 * 3 + 1];
      out_subx[m * 3 + 2] = p[i2 * 3 + 2];
    }
    last = i2;
    slot = nslot;
  }
}

// =====================================================================
// Kernel 3: edge aggregation. One wave32 per edge, one float4 feature
// chunk per lane. deg counts all in-edges; feature scatter only for
// sampled destinations (accumulates into L2-resident d_out region).
// =====================================================================
__global__ void agg_kernel(const int* __restrict__ src, const int* __restrict__ dst,
                           const float* __restrict__ feat, const int* __restrict__ inv,
                           float* __restrict__ deg, float* __restrict__ subfeat) {
  int gid = blockIdx.x * blockDim.x + threadIdx.x;
  int e = gid >> 5;
  int c = gid & 31;
  int s = 0, dv = 0, ld = -1;
  if (c == 0) {
    s  = src[e];
    dv = dst[e];
    atomicAdd(deg + dv, 1.0f);
    ld = inv[dv];
  }
  s  = __shfl(s, 0);
  ld = __shfl(ld, 0);
  if (ld >= 0) {
    const float4 v = reinterpret_cast<const float4*>(feat)[(size_t)s * 32 + c];
    float* o = subfeat + (size_t)ld * D_FEAT + c * 4;
    atomicAdd(o + 0, v.x);
    atomicAdd(o + 1, v.y);
    atomicAdd(o + 2, v.z);
    atomicAdd(o + 3, v.w);
  }
}

// =====================================================================
// Kernel 4: sub_feat /= max(deg, 1)
// =====================================================================
__global__ void finalize_kernel(const int* __restrict__ fpi, const float* __restrict__ deg,
                                float* __restrict__ subfeat) {
  int i = blockIdx.x * blockDim.x + threadIdx.x;
  int m = i >> 7;  // D_FEAT == 128
  int g = fpi[m];
  float scale = 1.0f / fmaxf(deg[g], 1.0f);
  subfeat[i] *= scale;
}

// =====================================================================
// Kernel 5: per-edge displacement, norm and mask on the induced subgraph
// =====================================================================
__global__ void edge_out_kernel(const int* __restrict__ src, const int* __restrict__ dst,
                                const int* __restrict__ inv, const float* __restrict__ subx,
                                float* __restrict__ od, float* __restrict__ ow,
                                float* __restrict__ om) {
  int e = blockIdx.x * blockDim.x + threadIdx.x;
  if (e >= N_EDGES) return;
  int ls = inv[src[e]];
  int ld = inv[dst[e]];
  bool mk = (ls >= 0) && (ld >= 0);
  float dx = 0.0f, dy = 0.0f, dz = 0.0f;
  if (mk) {
    dx = subx[ld * 3 + 0] - subx[ls * 3 + 0];
    dy = subx[ld * 3 + 1] - subx[ls * 3 + 1];
    dz = subx[ld * 3 + 2] - subx[ls * 3 + 2];
  }
  od[(size_t)e * 3 + 0] = dx;
  od[(size_t)e * 3 + 1] = dy;
  od[(size_t)e * 3 + 2] = dz;
  ow[e] = sqrtf(dx * dx + dy * dy + dz * dz);
  om[e] = mk ? 1.0f : 0.0f;
}

// =====================================================================
extern "C" void kernel_launch(void* const* d_in, const int* in_sizes, int n_in,
                              void* d_out, int out_size, void* d_ws, size_t ws_size,
                              hipStream_t stream) {
  const float* pos  = (const float*)d_in[0];   // [N,3]
  const float* feat = (const float*)d_in[1];   // [N,128]
  const int*   src  = (const int*)d_in[2];     // [E]
  const int*   dst  = (const int*)d_in[3];     // [E]
  (void)in_sizes; (void)n_in; (void)out_size; (void)ws_size;

  float* out         = (float*)d_out;
  float* out_subx    = out;                                   // M*3
  float* out_subfeat = out_subx + (size_t)M_SUB * 3;          // M*128
  float* out_d       = out_subfeat + (size_t)M_SUB * D_FEAT;  // E*3
  float* out_w       = out_d + (size_t)N_EDGES * 3;           // E
  float* out_mask    = out_w + N_EDGES;                       // E
  float* out_fp      = out_mask + N_EDGES;                    // M

  int*   inv = (int*)d_ws;                    // N ints
  float* deg = (float*)d_ws + N_NODES;        // N floats
  int*   fpi = (int*)d_ws + 2 * N_NODES;      // M ints

  pool_init_kernel<<<(M_SUB * D_FEAT + 255) / 256, 256, 0, stream>>>(inv, deg, out_subfeat);
  fps_kernel<<<BATCH, 1024, 0, stream>>>(pos, inv, fpi, out_subx, out_fp);
  agg_kernel<<<(N_EDGES * 32) / 256, 256, 0, stream>>>(src, dst, feat, inv, deg, out_subfeat);
  finalize_kernel<<<(M_SUB * D_FEAT) / 256, 256, 0, stream>>>(fpi, deg, out_subfeat);
  edge_out_kernel<<<(N_EDGES + 255) / 256, 256, 0, stream>>>(src, dst, inv, out_subx,
                                                             out_d, out_w, out_mask);
}